// BallQueryLayer_75015898792276
// MI455X (gfx1250) — compile-verified
//
#include <hip/hip_runtime.h>

// Ball query (first-K-by-index). 16x16 distance tiles via V_WMMA_F32_16X16X4_F32:
//   A row i  = [x1, y1, z1, 1]
//   B col j  = [-2x2, -2y2, -2z2, |p2_j|^2]
//   dot      = |p2_j|^2 - 2*p1.p2      compared vs  r^2 - |p1_i|^2
// 2 column tiles per iteration, back-to-back WMMAs, software-pipelined loads.

typedef __attribute__((ext_vector_type(2))) float v2f;
typedef __attribute__((ext_vector_type(8))) float v8f;

#define BQ_N1 8192
#define BQ_N2 16384
#define BQ_K 32
#define BQ_R2 (0.08f * 0.08f)

__global__ __launch_bounds__(256) void ballquery_wmma_kernel(
    const float* __restrict__ p1, const float* __restrict__ p2,
    int* __restrict__ mapping, int* __restrict__ numn,
    float* __restrict__ outp) {
  const int lane = threadIdx.x & 31;
  const int wave = threadIdx.x >> 5;
  const int rowBase = (blockIdx.x * (blockDim.x >> 5) + wave) << 4;  // 16 rows/wave
  if (rowBase >= BQ_N1) return;  // wave-uniform
  const int m = lane & 15;
  const bool hi = lane >= 16;

  // Every lane loads the query point for row (rowBase + m).
  const float qx = p1[(rowBase + m) * 3 + 0];
  const float qy = p1[(rowBase + m) * 3 + 1];
  const float qz = p1[(rowBase + m) * 3 + 2];
  const float threshVal = BQ_R2 - (qx * qx + qy * qy + qz * qz);

  // A fragment (16x4 f32): VGPR0 lanes0-15=K0, lanes16-31=K2; VGPR1: K1/K3.
  v2f a;
  a.x = hi ? qz : qx;
  a.y = hi ? 1.0f : qy;

  // Per-lane thresholds: D VGPR v holds row (v + 8*hi) in this lane-half.
  float th[8];
#pragma unroll
  for (int v = 0; v < 8; ++v) th[v] = __shfl(threshVal, v + (hi ? 8 : 0), 32);

  int cnt = hi ? BQ_K : 0;  // lanes 16..31 passive in append phase

  // Ballot per accumulator VGPR, pick this lane's row mask, ordered append.
  auto selectAppend = [&](const v8f& c, int jb) {
    unsigned msk[8];
#pragma unroll
    for (int v = 0; v < 8; ++v)
      msk[v] = __builtin_amdgcn_ballot_w32(c[v] <= th[v]);
    unsigned sel = msk[0];
#pragma unroll
    for (int v = 1; v < 8; ++v) sel = ((m & 7) == v) ? msk[v] : sel;
    unsigned rowmask = (m < 8) ? (sel & 0xFFFFu) : (sel >> 16);
    if (!hi) {
      while (rowmask && cnt < BQ_K) {
        const int nb = __builtin_ctz(rowmask);
        rowmask &= rowmask - 1;
        const int idx = jb + nb;
        const int slot = (rowBase + m) * BQ_K + cnt;
        mapping[slot] = idx;
        outp[slot * 3 + 0] = p2[idx * 3 + 0];
        outp[slot * 3 + 1] = p2[idx * 3 + 1];
        outp[slot * 3 + 2] = p2[idx * 3 + 2];
        ++cnt;
      }
    }
  };

  // Software pipeline: preload the first two 16-point tiles.
  float c0x = p2[(0 + m) * 3 + 0], c0y = p2[(0 + m) * 3 + 1], c0z = p2[(0 + m) * 3 + 2];
  float c1x = p2[(16 + m) * 3 + 0], c1y = p2[(16 + m) * 3 + 1], c1z = p2[(16 + m) * 3 + 2];

  for (int jb = 0; jb < BQ_N2; jb += 32) {
    // Issue next iteration's loads first (clamped; unused on final iteration).
    const int nj = (jb + 32 < BQ_N2) ? (jb + 32) : jb;
    const float n0x = p2[(nj + m) * 3 + 0];
    const float n0y = p2[(nj + m) * 3 + 1];
    const float n0z = p2[(nj + m) * 3 + 2];
    const float n1x = p2[(nj + 16 + m) * 3 + 0];
    const float n1y = p2[(nj + 16 + m) * 3 + 1];
    const float n1z = p2[(nj + 16 + m) * 3 + 2];
    const int pj = (jb + 64 < BQ_N2) ? (jb + 64) : jb;
    __builtin_prefetch(&p2[pj * 3], 0, 1);

    // B fragments for tiles jb and jb+16.
    v2f b0, b1;
    b0.x = hi ? (-2.0f * c0z) : (-2.0f * c0x);
    b0.y = hi ? (c0x * c0x + c0y * c0y + c0z * c0z) : (-2.0f * c0y);
    b1.x = hi ? (-2.0f * c1z) : (-2.0f * c1x);
    b1.y = hi ? (c1x * c1x + c1y * c1y + c1z * c1z) : (-2.0f * c1y);

    // Back-to-back WMMAs (independent accumulators -> XDL pipelining).
    v8f cc0 = {};
    v8f cc1 = {};
    cc0 = __builtin_amdgcn_wmma_f32_16x16x4_f32(false, a, false, b0, (short)0,
                                                cc0, false, false);
    cc1 = __builtin_amdgcn_wmma_f32_16x16x4_f32(false, a, false, b1, (short)0,
                                                cc1, false, false);

    // Ordered appends: tile jb first, then jb+16.
    selectAppend(cc0, jb);
    selectAppend(cc1, jb + 16);

    // Rotate pipeline registers.
    c0x = n0x; c0y = n0y; c0z = n0z;
    c1x = n1x; c1y = n1y; c1z = n1z;

    // Wave-uniform early exit once all 16 rows are full.
    const unsigned full = __builtin_amdgcn_ballot_w32(cnt >= BQ_K);
    if (full == 0xFFFFFFFFu) break;
  }

  if (!hi) {
    numn[rowBase + m] = cnt;
    for (int kk = cnt; kk < BQ_K; ++kk) {
      const int slot = (rowBase + m) * BQ_K + kk;
      mapping[slot] = 0;
      outp[slot * 3 + 0] = 0.0f;
      outp[slot * 3 + 1] = 0.0f;
      outp[slot * 3 + 2] = 0.0f;
    }
  }
}

extern "C" void kernel_launch(void* const* d_in, const int* in_sizes, int n_in,
                              void* d_out, int out_size, void* d_ws,
                              size_t ws_size, hipStream_t stream) {
  (void)in_sizes; (void)n_in; (void)d_ws; (void)ws_size; (void)out_size;
  const float* p1 = (const float*)d_in[0];  // [1, 8192, 3]
  const float* p2 = (const float*)d_in[1];  // [1, 16384, 3]

  // Output tuple, concatenated flat:
  //   mapping       int32 [1, 8192, 32]      -> elements [0, 262144)
  //   num_neighbors int32 [1, 8192]          -> elements [262144, 270336)
  //   outputs       f32   [1, 8192, 32, 3]   -> elements [270336, 1056768)
  int* mapping = (int*)d_out;
  int* numn = mapping + BQ_N1 * BQ_K;
  float* outp = (float*)d_out + BQ_N1 * BQ_K + BQ_N1;

  // 8 waves per block * 16 rows per wave = 128 rows per block.
  const int blocks = BQ_N1 / 128;  // 64
  ballquery_wmma_kernel<<<blocks, 256, 0, stream>>>(p1, p2, mapping, numn, outp);
}